// BatchedMessagePassing_89189290869442
// MI455X (gfx1250) — compile-verified
//
#include <hip/hip_runtime.h>

typedef float v2f __attribute__((ext_vector_type(2)));
typedef float v8f __attribute__((ext_vector_type(8)));

#define BATCH  64
#define NLAB   1000
#define DEPTH  10
#define WIDTH  100
#define ROWS   128               /* GEMM rows: r = b*2+k */
#define MT     8                 /* 128/16 row tiles */
#define KK     25                /* 100/4 K-steps of WMMA f32 16x16x4 */

// ---------- kernel 0: init successor-index scratch ----------
__global__ void init_tgt_kernel(int* __restrict__ tgt) {
    int i = blockIdx.x * blockDim.x + threadIdx.x;
    if (i < DEPTH * WIDTH) tgt[i] = -1;
}

// ---------- kernel 1: grid-wide scan of dense succ for nonzeros ----------
// (embarrassingly parallel -> many WGPs; also warms L2 with all of succ)
__global__ void scan_succ_kernel(const float* __restrict__ succ,
                                 int* __restrict__ tgt) {
    const int total = DEPTH * WIDTH * NLAB;
    for (int i = blockIdx.x * blockDim.x + threadIdx.x; i < total;
         i += gridDim.x * blockDim.x) {
        if (succ[i] != 0.0f) {
            int dw = i / NLAB;           // d*WIDTH + w
            tgt[dw] = i - dw * NLAB;     // n  (<=1 nonzero per source row)
        }
    }
}

// One 16x16 output tile of  msg += E(128x100) * Sd(100x1000).
// FULL=true  -> all 16 columns in range, fully branchless.
// FULL=false -> tail tile straddling col 1000, per-lane guarded.
template<bool FULL>
__device__ __forceinline__ void gemm_tile(const float* __restrict__ Sd,
                                          const float (*E)[WIDTH],
                                          float* __restrict__ msg,
                                          int nt, int mt, int half, int l15) {
    const int  col = nt * 16 + l15;
    const bool cok = FULL || (col < NLAB);
    // B fragments: columns of Sd, mirrored lane layout of A (ISA 7.12.2)
    v2f Bf[KK];
    #pragma unroll
    for (int kk = 0; kk < KK; ++kk) {
        int k0 = kk * 4 + half * 2;
        float b0 = cok ? Sd[(k0 + 0) * NLAB + col] : 0.0f;
        float b1 = cok ? Sd[(k0 + 1) * NLAB + col] : 0.0f;
        v2f bf = {b0, b1};
        Bf[kk] = bf;
    }
    const int arow = mt * 16 + l15;
    v8f acc;
    #pragma unroll
    for (int v = 0; v < 8; ++v) {        // C tile = current messages
        int row = mt * 16 + v + half * 8;
        acc[v] = cok ? msg[row * NLAB + col] : 0.0f;
    }
    #pragma unroll
    for (int kk = 0; kk < KK; ++kk) {
        int k0 = kk * 4 + half * 2;
        v2f Af = {E[arow][k0 + 0], E[arow][k0 + 1]};
        acc = __builtin_amdgcn_wmma_f32_16x16x4_f32(false, Af, false, Bf[kk],
                                                    (short)0, acc, false, false);
    }
    #pragma unroll
    for (int v = 0; v < 8; ++v) {
        int row = mt * 16 + v + half * 8;
        if (cok) msg[row * NLAB + col] = acc[v];
    }
}

// ---------- kernel 2: persistent single-WG serial message passing ----------
__launch_bounds__(1024)
__global__ void treecrf_mp_kernel(const float* __restrict__ em,
                                  const float* __restrict__ trans,
                                  const int*   __restrict__ layer_ids,
                                  const float* __restrict__ succ,
                                  const int*   __restrict__ tgt_in,
                                  float* __restrict__ msg) {
    __shared__ float E[ROWS][WIDTH];          // 51200 B
    __shared__ int   tgtAll[DEPTH * WIDTH];   //  4000 B
    __shared__ int   sMin, sMax;

    const int tid  = threadIdx.x;
    const int lane = tid & 31;
    const int wave = tid >> 5;          // 0..31 (wave32)
    const int half = (lane >> 4) & 1;
    const int l15  = lane & 15;

    // zero messages (harness poisons d_out); stage successor indices in LDS
    for (int i = tid; i < ROWS * NLAB; i += 1024) msg[i] = 0.0f;
    for (int i = tid; i < DEPTH * WIDTH; i += 1024) tgtAll[i] = tgt_in[i];
    __threadfence();
    __syncthreads();

    for (int d = 0; d < DEPTH; ++d) {
        const float* Sd  = succ + (size_t)d * WIDTH * NLAB;
        const int*   lid = layer_ids + d * WIDTH;
        const int*   tgt = tgtAll + d * WIDTH;

        // ---- active column window of Sd (exact: all nonzeros sit at tgt[w]) ----
        if (tid == 0) { sMin = 0x7fffffff; sMax = -1; }
        __syncthreads();
        if (tid < WIDTH) {
            int t = tgt[tid];
            if (t >= 0) { atomicMin(&sMin, t); atomicMax(&sMax, t); }
        }
        __syncthreads();
        const int bmin = sMin, bmax = sMax;   // workgroup-uniform
        if (bmax < 0) continue;               // no successors: messages unchanged

        // ---- Phase A: E[b*2+k][w] = lse_c( em[b,c,l_w]+msg[b,c,l_w]+T[tgt,l_w,k,c] )
        for (int i = tid; i < ROWS * WIDTH; i += 1024) {
            int w = i >> 7;            // i / 128
            int r = i & 127;           // b*2+k
            int b = r >> 1, k = r & 1;
            int t = tgt[w];
            float e = 0.0f;            // don't-care when S row is all-zero
            if (t >= 0) {
                int l = lid[w];
                float x0 = em[(b * 2 + 0) * NLAB + l] + msg[(b * 2 + 0) * NLAB + l];
                float x1 = em[(b * 2 + 1) * NLAB + l] + msg[(b * 2 + 1) * NLAB + l];
                const float* tp = trans + (((size_t)t * NLAB + l) * 2 + k) * 2;
                float a0 = x0 + tp[0];
                float a1 = x1 + tp[1];
                float m  = fmaxf(a0, a1);
                e = m + __logf(__expf(a0 - m) + __expf(a1 - m));
            }
            E[r][w] = e;
        }
        __syncthreads();

        // ---- Phase B: WMMA GEMM over active column tiles only ----
        const int t0 = bmin >> 4, t1 = bmax >> 4;
        const int ntiles = (t1 - t0 + 1) * MT;
        for (int ti = wave; ti < ntiles; ti += 32) {
            int nt = t0 + (ti >> 3);
            int mt = ti & 7;
            if (nt * 16 + 15 < NLAB)           // wave-uniform branch
                gemm_tile<true >(Sd, E, msg, nt, mt, half, l15);
            else
                gemm_tile<false>(Sd, E, msg, nt, mt, half, l15);
        }
        __threadfence();
        __syncthreads();
    }
}

extern "C" void kernel_launch(void* const* d_in, const int* in_sizes, int n_in,
                              void* d_out, int out_size, void* d_ws, size_t ws_size,
                              hipStream_t stream) {
    const float* emissions   = (const float*)d_in[0];
    const float* transitions = (const float*)d_in[1];
    const int*   layer_ids   = (const int*)d_in[2];
    const float* succ        = (const float*)d_in[3];
    float* out   = (float*)d_out;
    int*   wstgt = (int*)d_ws;          // 1000 ints = 4 KB of scratch
    (void)in_sizes; (void)n_in; (void)out_size; (void)ws_size;

    // 1) parallel: locate each node's successor from the dense 0/1 matrix
    init_tgt_kernel<<<1, 1024, 0, stream>>>(wstgt);
    scan_succ_kernel<<<512, 256, 0, stream>>>(succ, wstgt);
    // 2) serial: 10 dependent layers in one persistent workgroup
    treecrf_mp_kernel<<<1, 1024, 0, stream>>>(emissions, transitions, layer_ids,
                                              succ, wstgt, out);
}